// RNN_49065706390017
// MI455X (gfx1250) — compile-verified
//
#include <hip/hip_runtime.h>
#include <math.h>

// RNN (L independent tanh layers due to reference aliasing):
//   h[l]_t = tanh(x_t @ W_ih[l]^T + h[l]_{t-1} @ W_hh[l]^T + b_ih[l] + b_hh[l])
// out = (output[S,B,H] = h[1]_t, h_final[L,B,H]) concatenated.
//
// Per-step fused dual GEMM with V_WMMA_F32_16X16X4_F32.
// A-slab (16 B-rows) staged to LDS via gfx1250 async load-to-LDS, shared by
// all 8 waves; each wave register-blocks 2 N-tiles (2 v8f accumulators).

typedef float v2f __attribute__((ext_vector_type(2)));
typedef float v8f __attribute__((ext_vector_type(8)));
typedef int   v4i __attribute__((ext_vector_type(4)));

typedef __attribute__((address_space(1))) v4i* as1_v4i_ptr;
typedef __attribute__((address_space(3))) v4i* as3_v4i_ptr;

#define LL 2
#define BB 64
#define SS 512
#define II 1024
#define HH 1024
#define KC 64          // K-chunk staged in LDS (16 x 64 floats = 4 KB)
#define APAD 68        // padded LDS row stride (floats): 16 rows -> 16 distinct banks

#if defined(__has_builtin)
#if __has_builtin(__builtin_amdgcn_global_load_async_to_lds_b128)
#define HAVE_ASYNC_LDS 1
#endif
#endif

__device__ __forceinline__ void stage_wait_barrier()
{
#if defined(HAVE_ASYNC_LDS)
#if __has_builtin(__builtin_amdgcn_s_wait_asynccnt)
    __builtin_amdgcn_s_wait_asynccnt(0);
#else
    asm volatile("s_wait_asynccnt 0" ::: "memory");
#endif
#endif
    __syncthreads();
}

// One K=Kt GEMM accumulated into acc0/acc1.
//  Ablk: 16 x Kt row-major slab (row stride Astride), shared by the block.
//  B0/B1: W rows for this wave's two N-tiles.
__device__ __forceinline__ void gemm_part(
    float* __restrict__ Alds,
    const float* __restrict__ Ablk, int Astride,
    const float* __restrict__ B0,
    const float* __restrict__ B1,
    int Kt, v8f& acc0, v8f& acc1)
{
    const int t     = (int)threadIdx.x;
    const int sr    = t >> 4;              // staging row   0..15
    const int sc    = (t & 15) << 2;       // staging col   (floats, 16B aligned)
    const int lane  = t & 31;
    const int row16 = lane & 15;
    const int koff  = (lane >> 4) << 1;    // lanes 16-31 hold K=2,3 halves

    for (int kc = 0; kc < Kt; kc += KC) {
        __syncthreads();                   // previous chunk fully consumed
        {
            const float* gsrc = Ablk + (size_t)sr * Astride + kc + sc;
            float* ldst = Alds + sr * APAD + sc;
#if defined(HAVE_ASYNC_LDS)
            __builtin_amdgcn_global_load_async_to_lds_b128(
                (as1_v4i_ptr)gsrc, (as3_v4i_ptr)ldst, 0, 0);
#else
            *(float4*)ldst = *(const float4*)gsrc;
#endif
        }
        stage_wait_barrier();

        const float* arow = Alds + row16 * APAD + koff;
        const float* b0p  = B0 + kc + koff;
        const float* b1p  = B1 + kc + koff;
#pragma unroll 8
        for (int k = 0; k < KC; k += 4) {
            v2f a  = *(const v2f*)(arow + k);   // ds_load_b64
            v2f b0 = *(const v2f*)(b0p + k);    // global_load_b64
            v2f b1 = *(const v2f*)(b1p + k);
            acc0 = __builtin_amdgcn_wmma_f32_16x16x4_f32(
                false, a, false, b0, (short)0, acc0, false, false);
            acc1 = __builtin_amdgcn_wmma_f32_16x16x4_f32(
                false, a, false, b1, (short)0, acc1, false, false);
        }
    }
}

__global__ __launch_bounds__(256) void rnn_step_f32_wmma(
    const float* __restrict__ x_s,    // [B, I]   = x[s]
    const float* __restrict__ h_prev, // [L, B, H]
    float*       __restrict__ h_next, // [L, B, H] (ws ping-pong)
    const float* __restrict__ W_ih,   // [L, H, I]
    const float* __restrict__ W_hh,   // [L, H, H]
    const float* __restrict__ b_ih,   // [L, H]
    const float* __restrict__ b_hh,   // [L, H]
    float*       __restrict__ out,    // [S, B, H] + tail [L, B, H]
    int s)
{
    __shared__ float Alds[16 * APAD];

    const int lane = threadIdx.x & 31;
    const int wave = threadIdx.x >> 5;
    const int l    = blockIdx.y;
    const int tm   = blockIdx.x & 3;        // B tile (16 rows), fixed per block
    const int tg   = blockIdx.x >> 2;       // N group: 16 tn tiles per block
    const int tn0  = tg * 16 + wave * 2;    // this wave's two N tiles

    const int row16 = lane & 15;
    const int brow0 = tn0 * 16 + row16;          // W row for tile tn0
    const int brow1 = (tn0 + 1) * 16 + row16;    // W row for tile tn0+1

    v8f acc0 = {0.f,0.f,0.f,0.f,0.f,0.f,0.f,0.f};
    v8f acc1 = {0.f,0.f,0.f,0.f,0.f,0.f,0.f,0.f};

    // Input projection: K = I
    gemm_part(Alds,
              x_s + (size_t)(tm * 16) * II, II,
              W_ih + ((size_t)l * HH + brow0) * II,
              W_ih + ((size_t)l * HH + brow1) * II,
              II, acc0, acc1);
    // Recurrence: K = H
    gemm_part(Alds,
              h_prev + ((size_t)l * BB + tm * 16) * HH, HH,
              W_hh + ((size_t)l * HH + brow0) * HH,
              W_hh + ((size_t)l * HH + brow1) * HH,
              HH, acc0, acc1);

    // C/D layout: lanes 0-15 -> N=lane, M=r; lanes 16-31 -> N=lane-16, M=r+8.
    const int mbase = tm * 16 + ((lane >> 4) << 3);
    float* hdst = h_next + (size_t)l * BB * HH;
    float* odst = out + (size_t)s * BB * HH;
    float* fdst = out + (size_t)SS * BB * HH + (size_t)l * BB * HH;

#pragma unroll
    for (int t2 = 0; t2 < 2; ++t2) {
        const int g = (tn0 + t2) * 16 + (lane & 15);
        const float bias = b_ih[(size_t)l * HH + g] + b_hh[(size_t)l * HH + g];
        const v8f acc = t2 ? acc1 : acc0;
#pragma unroll
        for (int r = 0; r < 8; ++r) {
            const int bidx = mbase + r;
            const float v  = tanhf(acc[r] + bias);
            hdst[(size_t)bidx * HH + g] = v;
            if (l == LL - 1) odst[(size_t)bidx * HH + g] = v;
            if (s == SS - 1) fdst[(size_t)bidx * HH + g] = v;
        }
    }
}

extern "C" void kernel_launch(void* const* d_in, const int* in_sizes, int n_in,
                              void* d_out, int out_size, void* d_ws, size_t ws_size,
                              hipStream_t stream)
{
    (void)in_sizes; (void)n_in; (void)out_size; (void)ws_size;

    const float* x    = (const float*)d_in[0]; // [S, B, I]
    const float* h0   = (const float*)d_in[1]; // [L, B, H]
    const float* W_ih = (const float*)d_in[2]; // [L, H, I]
    const float* W_hh = (const float*)d_in[3]; // [L, H, H]
    const float* b_ih = (const float*)d_in[4]; // [L, H]
    const float* b_hh = (const float*)d_in[5]; // [L, H]
    float* out = (float*)d_out;

    float* hbuf0 = (float*)d_ws;                 // [L, B, H]
    float* hbuf1 = hbuf0 + (size_t)LL * BB * HH; // [L, B, H]

    dim3 grid(16, LL);
    dim3 block(256);

    const float* hp = h0;
    for (int s = 0; s < SS; ++s) {
        float* hn = (s & 1) ? hbuf1 : hbuf0;
        rnn_step_f32_wmma<<<grid, block, 0, stream>>>(
            x + (size_t)s * BB * II, hp, hn, W_ih, W_hh, b_ih, b_hh, out, s);
        hp = hn;
    }
}